// SGEMetric_8220567405269
// MI455X (gfx1250) — compile-verified
//
#include <hip/hip_runtime.h>
#include <math.h>

#define EPS_R    0.05f
#define INV_EPS  20.0f
#define N_ITER   50
#define PDIM     8
#define DFEAT    128
#define EMB      256

typedef __attribute__((ext_vector_type(2))) float v2f;
typedef __attribute__((ext_vector_type(8))) float v8f;

// ---------------- zero AGG / DEG ----------------
__global__ void zero_kernel(float* __restrict__ agg, float* __restrict__ deg, int n_nodes) {
    size_t stride = (size_t)gridDim.x * blockDim.x;
    size_t i = (size_t)blockIdx.x * blockDim.x + threadIdx.x;
    size_t total = (size_t)n_nodes * EMB;
    for (size_t j = i; j < total; j += stride) agg[j] = 0.0f;
    for (size_t j = i; j < (size_t)n_nodes; j += stride) deg[j] = 0.0f;
}

// ---------------- GEMM: H = X[M,128] * W[128,256], fp32 WMMA ----------------
// Block = 128 threads (4 waves). Each wave computes a 16(M) x 64(N) tile,
// four 16x16 accumulators fed by V_WMMA_F32_16X16X4_F32 (K stepped by 4).
__global__ void gemm_wmma(const float* __restrict__ X, const float* __restrict__ W,
                          float* __restrict__ H, int M) {
    int lane = threadIdx.x & 31;
    int wid  = threadIdx.x >> 5;        // n-group 0..3 (64 cols each)
    int mt   = blockIdx.x;              // 16-row tile
    if (mt * 16 >= M) return;

    int lo  = lane & 15;
    int hi  = lane >> 4;                // 0: K pair {k,k+1}, 1: {k+2,k+3}
    int row = mt * 16 + lo;             // A-matrix row for this lane
    int nb  = wid * 64 + lo;            // base column

    v8f c0 = {}, c1 = {}, c2 = {}, c3 = {};
    const float* xrow = X + (size_t)row * DFEAT;

#pragma unroll 4
    for (int k = 0; k < DFEAT; k += 4) {
        int kb = k + (hi << 1);
        v2f a = *(const v2f*)(xrow + kb);
        const float* wp = W + (size_t)kb * EMB + nb;
        v2f b0 = { wp[0],  wp[EMB]      };
        v2f b1 = { wp[16], wp[EMB + 16] };
        v2f b2 = { wp[32], wp[EMB + 32] };
        v2f b3 = { wp[48], wp[EMB + 48] };
        c0 = __builtin_amdgcn_wmma_f32_16x16x4_f32(false, a, false, b0, (short)0, c0, false, false);
        c1 = __builtin_amdgcn_wmma_f32_16x16x4_f32(false, a, false, b1, (short)0, c1, false, false);
        c2 = __builtin_amdgcn_wmma_f32_16x16x4_f32(false, a, false, b2, (short)0, c2, false, false);
        c3 = __builtin_amdgcn_wmma_f32_16x16x4_f32(false, a, false, b3, (short)0, c3, false, false);
    }

    int crow = mt * 16 + hi * 8;
#pragma unroll
    for (int r = 0; r < 8; ++r) {
        if (crow + r < M) {
            float* hp = H + (size_t)(crow + r) * EMB + nb;
            hp[0]  = c0[r];
            hp[16] = c1[r];
            hp[32] = c2[r];
            hp[48] = c3[r];
        }
    }
}

// ---------------- edge scatter: agg[dst] += h[src], deg[dst] += 1 ----------------
__global__ void scatter_kernel(const float* __restrict__ H, const int* __restrict__ ei,
                               float* __restrict__ agg, float* __restrict__ deg, int n_edges) {
    int w    = (int)(((size_t)blockIdx.x * blockDim.x + threadIdx.x) >> 5);
    int lane = threadIdx.x & 31;
    if (w >= n_edges) return;
    int src = ei[w];
    int dst = ei[n_edges + w];
    const float* hs = H + (size_t)src * EMB;
    float* ad = agg + (size_t)dst * EMB;
#pragma unroll
    for (int k = 0; k < EMB / 32; ++k) {
        int d = lane + 32 * k;
        atomicAdd(ad + d, hs[d]);
    }
    if (lane == 0) atomicAdd(deg + dst, 1.0f);
}

// ---------------- z = relu(h + agg / max(deg,1)) in place over H ----------------
__global__ void combine_kernel(float* __restrict__ H, const float* __restrict__ agg,
                               const float* __restrict__ deg, int n_nodes) {
    size_t stride = (size_t)gridDim.x * blockDim.x;
    size_t total  = (size_t)n_nodes * EMB;
    for (size_t j = (size_t)blockIdx.x * blockDim.x + threadIdx.x; j < total; j += stride) {
        float dg = fmaxf(deg[j >> 8], 1.0f);
        float v  = H[j] + agg[j] / dg;
        H[j] = v > 0.0f ? v : 0.0f;
    }
}

// ---------------- wave32 helpers ----------------
__device__ __forceinline__ float wave_sum32(float x) {
#pragma unroll
    for (int s = 16; s > 0; s >>= 1) x += __shfl_xor(x, s, 32);
    return x;
}
// uniform broadcast of lane l's value (v_readlane_b32 -> SGPR scalar operand)
__device__ __forceinline__ float lane_bcast(float v, int l) {
    return __int_as_float(__builtin_amdgcn_readlane(__float_as_int(v), l));
}

// ---------------- one OT problem (32 points x 8 dims each side) per wave ----------
// c[i]  = C[i][lane]  (column-per-lane)  -> in-lane g-update
// ct[j] = C[lane][j]  (row-per-lane)     -> in-lane f-update
// Cross-lane traffic per iteration: only v_readlane broadcasts of f and g.
template <bool SYM>
__device__ __forceinline__ float sinkhorn_ot(const float* pa, const float* pb) {
    const float LOG32 = 3.46573590280f;  // log(32)

    float c[32], ct[32];
#pragma unroll
    for (int i = 0; i < 32; ++i) {
        float s = 1e-12f;
#pragma unroll
        for (int d = 0; d < PDIM; ++d) {
            float ad = lane_bcast(pa[d], i);   // Xa[i][d] (uniform)
            float t  = ad - pb[d];
            s = fmaf(t, t, s);
        }
        c[i] = sqrtf(s);                        // p=1 cost: Euclidean distance
    }
    if (SYM) {
#pragma unroll
        for (int j = 0; j < 32; ++j) ct[j] = c[j];   // C symmetric
    } else {
#pragma unroll
        for (int j = 0; j < 32; ++j) {
            float s = 1e-12f;
#pragma unroll
            for (int d = 0; d < PDIM; ++d) {
                float bd = lane_bcast(pb[d], j);  // Xb[j][d] (uniform)
                float t  = pa[d] - bd;
                s = fmaf(t, t, s);
            }
            ct[j] = sqrtf(s);
        }
    }

    float fl = 0.0f;   // lane i holds f_i
    float gl = 0.0f;   // lane j holds g_j
#pragma unroll 1
    for (int it = 0; it < N_ITER; ++it) {
        // f_i = -eps * [ LSE_j((g_j - C_ij)/eps) - log32 ]   (in-lane over ct[])
        {
            float m = lane_bcast(gl, 0) - ct[0];
#pragma unroll
            for (int j = 1; j < 32; ++j) m = fmaxf(m, lane_bcast(gl, j) - ct[j]);
            float ss = 0.0f;
#pragma unroll
            for (int j = 0; j < 32; ++j)
                ss += __expf((lane_bcast(gl, j) - ct[j] - m) * INV_EPS);
            fl = -m - EPS_R * (__logf(ss) - LOG32);
        }
        // g_j = -eps * [ LSE_i((f_i - C_ij)/eps) - log32 ]   (in-lane over c[])
        {
            float m = lane_bcast(fl, 0) - c[0];
#pragma unroll
            for (int i = 1; i < 32; ++i) m = fmaxf(m, lane_bcast(fl, i) - c[i]);
            float ss = 0.0f;
#pragma unroll
            for (int i = 0; i < 32; ++i)
                ss += __expf((lane_bcast(fl, i) - c[i] - m) * INV_EPS);
            gl = -m - EPS_R * (__logf(ss) - LOG32);
        }
    }

    // <mu,f> + <nu,g> with uniform weights = mean(f) + mean(g)
    float tot = wave_sum32(fl + gl);
    return tot * (1.0f / 32.0f);
}

// ---------------- one pair (3 OT problems) per wave ----------------
__global__ void sinkhorn_kernel(const float* __restrict__ Z, const int* __restrict__ pairs,
                                const float* __restrict__ gdist, float* __restrict__ dist,
                                int n_pairs) {
    int w    = (int)(((size_t)blockIdx.x * blockDim.x + threadIdx.x) >> 5);
    int lane = threadIdx.x & 31;
    if (w >= n_pairs) return;
    int ia = pairs[w];
    int ib = pairs[n_pairs + w];

    float a[PDIM], b[PDIM];
    {
        const float4* p4 = (const float4*)(Z + (size_t)ia * EMB + lane * PDIM);
        float4 t0 = p4[0], t1 = p4[1];
        a[0]=t0.x; a[1]=t0.y; a[2]=t0.z; a[3]=t0.w;
        a[4]=t1.x; a[5]=t1.y; a[6]=t1.z; a[7]=t1.w;
    }
    {
        const float4* p4 = (const float4*)(Z + (size_t)ib * EMB + lane * PDIM);
        float4 t0 = p4[0], t1 = p4[1];
        b[0]=t0.x; b[1]=t0.y; b[2]=t0.z; b[3]=t0.w;
        b[4]=t1.x; b[5]=t1.y; b[6]=t1.z; b[7]=t1.w;
    }

    float ot_ab = sinkhorn_ot<false>(a, b);
    float ot_aa = sinkhorn_ot<true>(a, a);
    float ot_bb = sinkhorn_ot<true>(b, b);

    if (lane == 0) {
        float d  = ot_ab - 0.5f * (ot_aa + ot_bb);
        float gd = gdist[w];
        dist[w] = fabsf(d - gd) / gd;
    }
}

// ---------------- deterministic fixed-tree mean reduction ----------------
__global__ void reduce_kernel(const float* __restrict__ dist, float* __restrict__ out, int n) {
    __shared__ float sh[256];
    int t = threadIdx.x;
    float s = 0.0f;
    for (int j = t; j < n; j += 256) s += dist[j];
    sh[t] = s;
    __syncthreads();
    for (int k = 128; k > 0; k >>= 1) {
        if (t < k) sh[t] += sh[t + k];
        __syncthreads();
    }
    if (t == 0) out[0] = sh[0] / (float)n;
}

extern "C" void kernel_launch(void* const* d_in, const int* in_sizes, int n_in,
                              void* d_out, int out_size, void* d_ws, size_t ws_size,
                              hipStream_t stream) {
    const float* x     = (const float*)d_in[0];   // [n_nodes, 128]
    const float* W     = (const float*)d_in[1];   // [128, 256]
    const int*   ei    = (const int*)d_in[2];     // [2, n_edges]
    const int*   pairs = (const int*)d_in[3];     // [2, n_pairs]
    const float* gd    = (const float*)d_in[4];   // [n_pairs]
    float* out = (float*)d_out;

    int n_nodes = in_sizes[0] / DFEAT;
    int n_edges = in_sizes[2] / 2;
    int n_pairs = in_sizes[4];

    float* H    = (float*)d_ws;                         // [n_nodes, 256]
    float* AGG  = H   + (size_t)n_nodes * EMB;          // [n_nodes, 256]
    float* DEG  = AGG + (size_t)n_nodes * EMB;          // [n_nodes]
    float* DIST = DEG + n_nodes;                        // [n_pairs]

    zero_kernel<<<2048, 256, 0, stream>>>(AGG, DEG, n_nodes);
    gemm_wmma<<<(n_nodes + 15) / 16, 128, 0, stream>>>(x, W, H, n_nodes);
    {
        size_t threads = (size_t)n_edges * 32;
        scatter_kernel<<<(unsigned)((threads + 255) / 256), 256, 0, stream>>>(H, ei, AGG, DEG, n_edges);
    }
    combine_kernel<<<2048, 256, 0, stream>>>(H, AGG, DEG, n_nodes);
    {
        size_t threads = (size_t)n_pairs * 32;
        sinkhorn_kernel<<<(unsigned)((threads + 127) / 128), 128, 0, stream>>>(H, pairs, gd, DIST, n_pairs);
    }
    reduce_kernel<<<1, 256, 0, stream>>>(DIST, out, n_pairs);
}